// KernelEstimator_35407710388315
// MI455X (gfx1250) — compile-verified
//
#include <hip/hip_runtime.h>

#define KW 7
#define TAPS 49
#define RCOLS 52         // raw row stride (keeps WMMA A-loads 8B aligned)
#define CHUNKS 12        // WMMA sums taps 0..47; tap 48 added per-lane in regs
#define WAVES 8
#define BLOCK 256

typedef __attribute__((ext_vector_type(2))) float v2f;
typedef __attribute__((ext_vector_type(8))) float v8f;

__device__ __forceinline__ float cubic_contrib(float x) {
    // a = -0.5 bicubic
    float ax  = __builtin_fabsf(x);
    float ax2 = ax * ax;
    float ax3 = ax2 * ax;
    float p1 = fmaf(1.5f, ax3, fmaf(-2.5f, ax2, 1.0f));                    // |x| <= 1
    float p2 = fmaf(-0.5f, ax3, fmaf(2.5f, ax2, fmaf(-4.0f, ax, 2.0f)));   // 1 < |x| <= 2
    return (ax <= 1.0f) ? p1 : ((ax <= 2.0f) ? p2 : 0.0f);
}

__global__ __launch_bounds__(BLOCK) void KernelEstimator_35407710388315_kernel(
    const float* __restrict__ m, const float* __restrict__ g,
    const int* __restrict__ Wp, const int* __restrict__ yi,
    float* __restrict__ out, int N)
{
    // One slab per wave: raw weights at stride-52 rows, later overwritten by
    // packed (stride-49) normalized weights for the flat copy-out.
    __shared__ float sW[WAVES][32 * RCOLS];
    __shared__ float sSum[WAVES][32];

    const int tid   = threadIdx.x;
    const int wv    = tid >> 5;
    const int ln    = tid & 31;
    const int wbase = blockIdx.x * BLOCK + wv * 32;
    const int i     = wbase + ln;
    const int ic    = (i < N) ? i : (N - 1);   // clamp: EXEC stays full for WMMA

    const unsigned Wimg = (unsigned)Wp[0];

    const float m00 = m[0], m01 = m[1], m02 = m[2];
    const float m10 = m[3], m11 = m[4], m12 = m[5];
    const float m20 = m[6], m21 = m[7], m22 = m[8];

    const unsigned j   = (unsigned)yi[ic];
    const unsigned yyi = j / Wimg;
    const unsigned xxi = j - yyi * Wimg;
    const float xx = (float)xxi, yv = (float)yyi;

    auto proj = [&](float px, float py, float& ox, float& oy) {
        float X = fmaf(m00, px, fmaf(m01, py, m02));
        float Y = fmaf(m10, px, fmaf(m11, py, m12));
        float Z = fmaf(m20, px, fmaf(m21, py, m22));
        float rz = 1.0f / Z;
        ox = X * rz; oy = Y * rz;
    };

    float plx, ply, prx, pry, ptx, pty, pbx, pby;
    proj(xx - 0.5f, yv, plx, ply);
    proj(xx + 0.5f, yv, prx, pry);
    proj(xx, yv - 0.5f, ptx, pty);
    proj(xx, yv + 0.5f, pbx, pby);

    float du0 = pbx - ptx, du1 = pby - pty;
    float dv0 = prx - plx, dv1 = pry - ply;

    float len_du = sqrtf(fmaf(du0, du0, du1 * du1));
    if (len_du < 1.0f) { float r = 1.0f / len_du; du0 *= r; du1 *= r; len_du = 1.0f; }
    float len_dv = sqrtf(fmaf(dv0, dv0, dv1 * dv1));
    if (len_dv < 1.0f) { float r = 1.0f / len_dv; dv0 *= r; dv1 *= r; len_dv = 1.0f; }

    float det = fmaf(du0, dv1, -(du1 * dv0));
    float ilu = 1.0f / (len_du * det);
    float ilv = 1.0f / (len_dv * det);

    const float Axc = du0 * dv1 * ilu - du1 * dv0 * ilv;
    const float Bxc = du0 * du1 * (ilv - ilu);
    const float Ayc = dv0 * dv1 * (ilu - ilv);
    const float Byc = du0 * dv1 * ilv - du1 * dv0 * ilu;

    const float posx = g[ic]     + 0.5f;
    const float posy = g[N + ic] + 0.5f;
    const float fx = posx - floorf(posx);
    const float fy = posy - floorf(posy);

    // ---- phase 1: weights into registers + raw copy to LDS (stride 52) ----
    float wreg[TAPS];
    #pragma unroll
    for (int ti = 0; ti < KW; ++ti) {
        const float wyv = fy + 2.5f - (float)ti;
        const float bxw = Bxc * wyv;
        const float byw = Byc * wyv;
        #pragma unroll
        for (int tj = 0; tj < KW; ++tj) {
            const float wxv  = fx + 2.5f - (float)tj;
            const float xval = fmaf(Axc, wxv, bxw);
            const float yval = fmaf(Ayc, wxv, byw);
            wreg[ti * KW + tj] = cubic_contrib(xval) * cubic_contrib(yval);
        }
    }
    {
        float* rowp = &sW[wv][ln * RCOLS];
        #pragma unroll
        for (int k = 0; k < TAPS; ++k) rowp[k] = wreg[k];
    }
    __syncthreads();

    // ---- phase 2: fp32 partial row-sums (taps 0..47) via WMMA f32 16x16x4 ----
    // A layout: lanes 0-15 hold K=0,1 of row M=lane; lanes 16-31 hold K=2,3.
    // B = ones -> D[M][*] = sum_k A[M][k]; C chained over 12 K=4 chunks.
    const int row = ln & 15;
    const int kb  = (ln >> 4) << 1;
    #pragma unroll
    for (int h = 0; h < 2; ++h) {
        v8f acc = {0.f, 0.f, 0.f, 0.f, 0.f, 0.f, 0.f, 0.f};
        const float* src = &sW[wv][(h * 16 + row) * RCOLS];
        #pragma unroll
        for (int c = 0; c < CHUNKS; ++c) {
            v2f a; a.x = src[4 * c + kb]; a.y = src[4 * c + kb + 1];
            v2f b; b.x = 1.0f; b.y = 1.0f;
            acc = __builtin_amdgcn_wmma_f32_16x16x4_f32(
                false, a, false, b, (short)0, acc, false, false);
        }
        // D layout: lanes 0-15 VGPR r => M=r ; lanes 16-31 VGPR r => M=8+r
        if ((ln & 15) == 0) {
            const int pbase = h * 16 + ((ln >> 4) << 3);
            #pragma unroll
            for (int r = 0; r < 8; ++r) sSum[wv][pbase + r] = acc[r];
        }
    }
    __syncthreads();

    // ---- phase 3: wave-parallel reciprocal, normalize in regs,
    //               rewrite slab as packed stride-49 rows ----
    const float inv = 1.0f / (sSum[wv][ln] + wreg[48]);
    {
        float* packp = &sW[wv][ln * TAPS];
        #pragma unroll
        for (int k = 0; k < TAPS; ++k) packp[k] = wreg[k] * inv;
    }
    __syncthreads();

    // ---- phase 4: flat LDS -> global float4 memcpy (fully coalesced) ----
    const long long total = (long long)N * TAPS;
    const long long obase = (long long)wbase * TAPS;
    const float* flat = &sW[wv][0];
    #pragma unroll 1
    for (int c = 0; c < 13; ++c) {
        const int q = c * 32 + ln;              // float4 index in wave region
        if (q < 392) {                          // 392 * 4 = 32 * 49 dwords
            const int t0 = q * 4;
            const long long oi = obase + (long long)t0;
            const float4 v4 = *(const float4*)(flat + t0);
            if (oi + 4 <= total) {
                *(float4*)(out + oi) = v4;
            } else {                            // ragged tail of last wave
                const float vv[4] = {v4.x, v4.y, v4.z, v4.w};
                #pragma unroll
                for (int d = 0; d < 4; ++d)
                    if (oi + (long long)d < total) out[oi + d] = vv[d];
            }
        }
    }
}

extern "C" void kernel_launch(void* const* d_in, const int* in_sizes, int n_in,
                              void* d_out, int out_size, void* d_ws, size_t ws_size,
                              hipStream_t stream) {
    const float* m  = (const float*)d_in[0];   // (3,3) fp32
    const float* g  = (const float*)d_in[1];   // (2,N) fp32
    const int*   Wp = (const int*)d_in[3];     // scalar W
    const int*   yi = (const int*)d_in[4];     // (N,) int32
    float*       o  = (float*)d_out;           // (N,49) fp32
    const int N = in_sizes[4];
    const int blocks = (N + BLOCK - 1) / BLOCK;
    hipLaunchKernelGGL(KernelEstimator_35407710388315_kernel,
                       dim3(blocks), dim3(BLOCK), 0, stream,
                       m, g, Wp, yi, o, N);
}